// boxes_and_scores_merger_77841987272769
// MI455X (gfx1250) — compile-verified
//
#include <hip/hip_runtime.h>
#include <hip/hip_bf16.h>
#include <stdint.h>

#define N_YOLO 8400
#define N_DETR 300
#define N_TOT  8700
#define NPAD   8704          // N_TOT rounded up to 256
#define NBLK   272           // NPAD / 32 mask words per row
#define NSORT  16384         // power-of-two pad for bitonic sort
#define TILE   256           // columns staged per LDS tile in mask kernel

// ---------------- workspace layout (all offsets 256B aligned) ----------------
// unsorted SoA x1,y1,x2,y2 : 4*NPAD floats
// unsorted conf            : NPAD floats
// order (sorted -> orig)   : NPAD u32
// sorted SoA x1,y1,x2,y2   : 4*NPAD floats
// sorted conf              : NPAD floats
// keep flags               : NPAD u32
// suppression mask         : NPAD * NBLK u32
static constexpr size_t OFF_UX    = 0;
static constexpr size_t OFF_UCONF = OFF_UX    + (size_t)4 * NPAD * 4;
static constexpr size_t OFF_ORDER = OFF_UCONF + (size_t)NPAD * 4;
static constexpr size_t OFF_SX    = OFF_ORDER + (size_t)NPAD * 4;
static constexpr size_t OFF_SCONF = OFF_SX    + (size_t)4 * NPAD * 4;
static constexpr size_t OFF_KEEP  = OFF_SCONF + (size_t)NPAD * 4;
static constexpr size_t OFF_MASK  = OFF_KEEP  + (size_t)NPAD * 4;

// ---------------------------------------------------------------------------
// 1) clip conf, cxcywh -> xyxy, write unsorted SoA
__global__ void __launch_bounds__(256) k_prep(const float* __restrict__ yb,
                                              const float* __restrict__ yc,
                                              const float* __restrict__ rb,
                                              const float* __restrict__ rc,
                                              float* __restrict__ ux,
                                              float* __restrict__ uconf) {
  int i = blockIdx.x * 256 + threadIdx.x;
  if (i >= N_TOT) return;
  float cx, cy, w, h, c;
  if (i < N_YOLO) {
    const float* b = yb + (size_t)i * 4;
    cx = b[0]; cy = b[1]; w = b[2]; h = b[3];
    c = yc[i];
  } else {
    const float* b = rb + (size_t)(i - N_YOLO) * 4;
    cx = b[0]; cy = b[1]; w = b[2]; h = b[3];
    c = rc[i - N_YOLO];
  }
  c = fminf(fmaxf(c, 0.0f), 1.0f);
  ux[0 * NPAD + i] = cx - 0.5f * w;
  ux[1 * NPAD + i] = cy - 0.5f * h;
  ux[2 * NPAD + i] = cx + 0.5f * w;
  ux[3 * NPAD + i] = cy + 0.5f * h;
  uconf[i] = c;
}

// ---------------------------------------------------------------------------
// 2) single-workgroup bitonic sort of 16384 64-bit keys in 128KB LDS (CDNA5
//    only: needs > 64KB LDS per workgroup). key = ((~confbits)<<32)|idx so an
//    ascending sort reproduces stable argsort(-conf) (conf desc, idx asc).
extern __shared__ unsigned long long skey[];
__global__ void __launch_bounds__(1024) k_sort(const float* __restrict__ uconf,
                                               unsigned int* __restrict__ order) {
  for (int i = threadIdx.x; i < NSORT; i += 1024) {
    unsigned long long key;
    if (i < N_TOT) {
      unsigned int bits = __float_as_uint(uconf[i]);   // conf >= 0 -> monotonic bits
      key = (((unsigned long long)(0xFFFFFFFFu - bits)) << 32) | (unsigned int)i;
    } else {
      key = 0xFFFFFFFFFFFFFFFFull;                     // pad sorts last
    }
    skey[i] = key;
  }
  __syncthreads();
  for (int k = 2; k <= NSORT; k <<= 1) {
    for (int j = k >> 1; j > 0; j >>= 1) {
      for (int i = threadIdx.x; i < NSORT; i += 1024) {
        int ixj = i ^ j;
        if (ixj > i) {
          unsigned long long a = skey[i], b = skey[ixj];
          bool up = ((i & k) == 0);
          if (up ? (a > b) : (a < b)) { skey[i] = b; skey[ixj] = a; }
        }
      }
      __syncthreads();
    }
  }
  for (int r = threadIdx.x; r < N_TOT; r += 1024)
    order[r] = (unsigned int)(skey[r] & 0xFFFFFFFFull);
}

// ---------------------------------------------------------------------------
// 3) gather sorted SoA + conf; zero the pad tail
__global__ void __launch_bounds__(256) k_gather(const unsigned int* __restrict__ order,
                                                const float* __restrict__ ux,
                                                const float* __restrict__ uconf,
                                                float* __restrict__ sx,
                                                float* __restrict__ sconf) {
  int r = blockIdx.x * 256 + threadIdx.x;
  if (r >= NPAD) return;
  if (r < N_TOT) {
    unsigned int o = order[r];
    sx[0 * NPAD + r] = ux[0 * NPAD + o];
    sx[1 * NPAD + r] = ux[1 * NPAD + o];
    sx[2 * NPAD + r] = ux[2 * NPAD + o];
    sx[3 * NPAD + r] = ux[3 * NPAD + o];
    sconf[r] = uconf[o];
  } else {
    sx[0 * NPAD + r] = 0.0f; sx[1 * NPAD + r] = 0.0f;
    sx[2 * NPAD + r] = 0.0f; sx[3 * NPAD + r] = 0.0f;
    sconf[r] = 0.0f;
  }
}

// ---------------------------------------------------------------------------
// 4) pairwise suppression bitmask. Each thread owns one row; 4KB column tiles
//    (x1,y1,x2,y2 x 256 cols) are staged into LDS with CDNA5 async tensor-path
//    loads: one global_load_async_to_lds_b128 per lane, ASYNCcnt-tracked.
__global__ void __launch_bounds__(256) k_mask(const float* __restrict__ ssoa,
                                              unsigned int* __restrict__ mask) {
  __shared__ float tile[4 * TILE];   // lx1[256] ly1[256] lx2[256] ly2[256]
  const int row = blockIdx.x * 256 + threadIdx.x;
  const float rx1 = ssoa[0 * NPAD + row];
  const float ry1 = ssoa[1 * NPAD + row];
  const float rx2 = ssoa[2 * NPAD + row];
  const float ry2 = ssoa[3 * NPAD + row];
  const float rarea = (rx2 - rx1) * (ry2 - ry1);

  // raw LDS byte offset of tile[]: low 32 bits of the flat address
  const unsigned int lds_base = (unsigned int)(uintptr_t)(&tile[0]);
  const int a = threadIdx.x >> 6;          // which of the 4 arrays this lane fills
  const int e = (threadIdx.x & 63) << 2;   // element within array (4 floats / lane)
  const unsigned int ldsb = lds_base + (unsigned int)((a * TILE + e) * 4);
  const unsigned long long sbase = (unsigned long long)(uintptr_t)ssoa;

  for (int t = 0; t < NPAD / TILE; ++t) {
    const int col0 = t * TILE;
    const unsigned int gofs = (unsigned int)(((size_t)a * NPAD + col0 + e) * sizeof(float));
    // CDNA5 async global->LDS (GVS mode: saddr base + 32-bit vaddr offset)
    asm volatile("global_load_async_to_lds_b128 %0, %1, %2"
                 :: "v"(ldsb), "v"(gofs), "s"(sbase) : "memory");
    asm volatile("s_wait_asynccnt 0x0" ::: "memory");
    __syncthreads();

    #pragma unroll
    for (int w = 0; w < TILE / 32; ++w) {
      unsigned int bits = 0u;
      for (int b = 0; b < 32; ++b) {
        const int jc = w * 32 + b;
        const float cx1 = tile[0 * TILE + jc];
        const float cy1 = tile[1 * TILE + jc];
        const float cx2 = tile[2 * TILE + jc];
        const float cy2 = tile[3 * TILE + jc];
        float iw = fminf(rx2, cx2) - fmaxf(rx1, cx1); iw = fmaxf(iw, 0.0f);
        float ih = fminf(ry2, cy2) - fmaxf(ry1, cy1); ih = fmaxf(ih, 0.0f);
        const float inter = iw * ih;
        const float uni = rarea + (cx2 - cx1) * (cy2 - cy1) - inter;
        const float iou = inter / fmaxf(uni, 1e-9f);     // mirror reference exactly
        const bool sup = (iou > 0.5f) && ((col0 + jc) > row);
        bits |= (sup ? 1u : 0u) << b;
      }
      mask[(size_t)row * NBLK + (size_t)(t * (TILE / 32) + w)] = bits;
    }
    __syncthreads();   // tile consumed; safe to overwrite next iteration
  }
}

// ---------------------------------------------------------------------------
// 5) sequential greedy scan (the inherently serial fori_loop of the reference)
__global__ void __launch_bounds__(256) k_scan(const unsigned int* __restrict__ mask,
                                              unsigned int* __restrict__ keep) {
  __shared__ unsigned int remv[NBLK];
  __shared__ int flag;
  for (int w = threadIdx.x; w < NBLK; w += 256) remv[w] = 0u;
  __syncthreads();
  for (int i = 0; i < N_TOT; ++i) {
    if (threadIdx.x == 0) {
      const int k = ((remv[i >> 5] >> (i & 31)) & 1u) ? 0 : 1;
      flag = k;
      keep[i] = (unsigned int)k;
    }
    __syncthreads();
    if (flag) {
      for (int w = threadIdx.x; w < NBLK; w += 256)
        remv[w] |= mask[(size_t)i * NBLK + w];
    }
    __syncthreads();
  }
}

// ---------------------------------------------------------------------------
// 6) write [x1,y1,x2,y2,conf] * keep
__global__ void __launch_bounds__(256) k_out(const float* __restrict__ sx,
                                             const float* __restrict__ sconf,
                                             const unsigned int* __restrict__ keep,
                                             float* __restrict__ out) {
  int r = blockIdx.x * 256 + threadIdx.x;
  if (r >= N_TOT) return;
  const float m = keep[r] ? 1.0f : 0.0f;
  out[(size_t)r * 5 + 0] = sx[0 * NPAD + r] * m;
  out[(size_t)r * 5 + 1] = sx[1 * NPAD + r] * m;
  out[(size_t)r * 5 + 2] = sx[2 * NPAD + r] * m;
  out[(size_t)r * 5 + 3] = sx[3 * NPAD + r] * m;
  out[(size_t)r * 5 + 4] = sconf[r] * m;
}

// ---------------------------------------------------------------------------
extern "C" void kernel_launch(void* const* d_in, const int* in_sizes, int n_in,
                              void* d_out, int out_size, void* d_ws, size_t ws_size,
                              hipStream_t stream) {
  (void)in_sizes; (void)n_in; (void)out_size; (void)ws_size;
  const float* yb = (const float*)d_in[0];   // yolo_cxcywh   (8400,4)
  const float* yc = (const float*)d_in[1];   // yolo conf     (8400)
  const float* rb = (const float*)d_in[2];   // rtdetr_cxcywh (300,4)
  const float* rc = (const float*)d_in[3];   // rtdetr conf   (300)

  char* ws = (char*)d_ws;
  float*        ux    = (float*)(ws + OFF_UX);
  float*        uconf = (float*)(ws + OFF_UCONF);
  unsigned int* order = (unsigned int*)(ws + OFF_ORDER);
  float*        sx    = (float*)(ws + OFF_SX);
  float*        sconf = (float*)(ws + OFF_SCONF);
  unsigned int* keep  = (unsigned int*)(ws + OFF_KEEP);
  unsigned int* mask  = (unsigned int*)(ws + OFF_MASK);

  const int nblk256 = NPAD / 256;  // 34

  k_prep  <<<nblk256, 256, 0, stream>>>(yb, yc, rb, rc, ux, uconf);
  k_sort  <<<1, 1024, NSORT * sizeof(unsigned long long), stream>>>(uconf, order);
  k_gather<<<nblk256, 256, 0, stream>>>(order, ux, uconf, sx, sconf);
  k_mask  <<<nblk256, 256, 0, stream>>>(sx, mask);
  k_scan  <<<1, 256, 0, stream>>>(mask, keep);
  k_out   <<<nblk256, 256, 0, stream>>>(sx, sconf, keep, (float*)d_out);
}